// NodeModel_38439957299909
// MI455X (gfx1250) — compile-verified
//
#include <hip/hip_runtime.h>

#define N_NODES 50000
#define N_EDGES 500000
#define D_IN    128
#define D_X     384            // 3 * D_IN
#define D_OUT   128
#define BN_EPS  1e-5f

typedef float v2f __attribute__((ext_vector_type(2)));
typedef float v8f __attribute__((ext_vector_type(8)));

// ---------------- workspace layout (in floats) ----------------
#define WS_RECV   0L
#define WS_SEND   (WS_RECV + (long)N_NODES * D_IN)     // 6,400,000
#define WS_H      (WS_SEND + (long)N_NODES * D_IN)     // 12,800,000
#define WS_SUM    (WS_H    + (long)N_NODES * D_OUT)    // 19,200,000
#define WS_SUMSQ  (WS_SUM   + 128)
#define WS_SCALE  (WS_SUMSQ + 128)
#define WS_SHIFT  (WS_SCALE + 128)

// ---------------- zero fill ----------------
__global__ void nm_zero(float* __restrict__ p, long n) {
    long i = (long)blockIdx.x * blockDim.x + threadIdx.x;
    long stride = (long)gridDim.x * blockDim.x;
    for (; i < n; i += stride) p[i] = 0.0f;
}

// ---------------- edge scatter: 32 lanes per edge, 4 channels/lane ----------------
__global__ void nm_scatter(const float* __restrict__ edge_attr,
                           const int* __restrict__ row,     // senders
                           const int* __restrict__ col,     // receivers
                           float* __restrict__ recv,
                           float* __restrict__ send) {
    long tid = (long)blockIdx.x * blockDim.x + threadIdx.x;
    long e = tid >> 5;
    int lane = (int)(tid & 31);
    if (e >= N_EDGES) return;
    int c0 = lane * 4;
    const float4 v = *(const float4*)(edge_attr + e * D_IN + c0);
    long r = col[e];   // receiver aggregation target
    long s = row[e];   // sender aggregation target
    float* pr = recv + r * D_IN + c0;
    float* ps = send + s * D_IN + c0;
    unsafeAtomicAdd(pr + 0, v.x);
    unsafeAtomicAdd(pr + 1, v.y);
    unsafeAtomicAdd(pr + 2, v.z);
    unsafeAtomicAdd(pr + 3, v.w);
    unsafeAtomicAdd(ps + 0, v.x);
    unsafeAtomicAdd(ps + 1, v.y);
    unsafeAtomicAdd(ps + 2, v.z);
    unsafeAtomicAdd(ps + 3, v.w);
}

// ---------------- fp32 WMMA GEMM + bias + ReLU, 5Mx2N register blocking ----
// h = relu(X @ W^T + b), X = [recv | send | node] (virtual concat).
// One wave owns a 80x32 super-tile: 5 M-tiles x 2 N-tiles, 10 v8f accumulators.
// Per K-step: 5 A-frag loads + 2 B-frag loads -> 10 x V_WMMA_F32_16X16X4_F32.
// Grid: 625 blocks x 4 waves; 625*80 == 50000 rows, 4*32 == 128 channels. No guards.
__global__ void nm_gemm_relu(const float* __restrict__ recv,
                             const float* __restrict__ send,
                             const float* __restrict__ node,
                             const float* __restrict__ W,     // [D_OUT, D_X] row-major
                             const float* __restrict__ bias,  // [D_OUT]
                             float* __restrict__ h) {         // [N_NODES, D_OUT]
    const int wave = threadIdx.x >> 5;             // 0..3  -> N-tile pair
    const int lane = threadIdx.x & 31;
    const int mgrp = blockIdx.x;                   // 0..624 -> 5 M-tiles
    const int mr   = lane & 15;
    const int half = lane >> 4;                    // 0: K={k,k+1}, 1: K={k+2,k+3}
    const int ch0  = wave * 32 + mr;               // first N-tile channel
    const int ch1  = ch0 + 16;                     // second N-tile channel

    // A-row pointers for the 5 M-tiles (row = mgrp*80 + t*16 + mr)
    const long row0 = (long)mgrp * 80 + mr;

    v8f acc[10];
    {
        const float bv0 = bias[ch0];
        const float bv1 = bias[ch1];
#pragma unroll
        for (int t = 0; t < 5; ++t) {
#pragma unroll
            for (int i = 0; i < 8; ++i) { acc[t * 2 + 0][i] = bv0; acc[t * 2 + 1][i] = bv1; }
        }
    }

    const float* segs[3] = {recv, send, node};
#pragma unroll
    for (int s = 0; s < 3; ++s) {
        const float* Xs  = segs[s] + row0 * D_IN + 2 * half;
        const float* Wr0 = W + (long)ch0 * D_X + s * D_IN + 2 * half;
        const float* Wr1 = W + (long)ch1 * D_X + s * D_IN + 2 * half;
#pragma unroll 2
        for (int k = 0; k < D_IN; k += 4) {
            v2f b0 = *(const v2f*)(Wr0 + k);
            v2f b1 = *(const v2f*)(Wr1 + k);
#pragma unroll
            for (int t = 0; t < 5; ++t) {
                v2f a = *(const v2f*)(Xs + (long)t * 16 * D_IN + k);
                acc[t * 2 + 0] = __builtin_amdgcn_wmma_f32_16x16x4_f32(
                    false, a, false, b0, (short)0, acc[t * 2 + 0], false, false);
                acc[t * 2 + 1] = __builtin_amdgcn_wmma_f32_16x16x4_f32(
                    false, a, false, b1, (short)0, acc[t * 2 + 1], false, false);
            }
        }
    }

    // ReLU + store. C/D layout: VGPR v -> row (v + 8*half), column = ch.
#pragma unroll
    for (int t = 0; t < 5; ++t) {
#pragma unroll
        for (int v = 0; v < 8; ++v) {
            long m = (long)mgrp * 80 + t * 16 + half * 8 + v;
            float a0 = acc[t * 2 + 0][v];
            float a1 = acc[t * 2 + 1][v];
            h[m * D_OUT + ch0] = a0 > 0.0f ? a0 : 0.0f;
            h[m * D_OUT + ch1] = a1 > 0.0f ? a1 : 0.0f;
        }
    }
}

// ---------------- per-channel sum / sumsq ----------------
__global__ void nm_bn_stats(const float* __restrict__ h,
                            float* __restrict__ sum,
                            float* __restrict__ sumsq) {
    const int c = threadIdx.x;                 // 0..127, blockDim.x == 128
    long r0 = (long)blockIdx.x * 128;
    long r1 = r0 + 128;
    if (r1 > N_NODES) r1 = N_NODES;
    float s = 0.0f, s2 = 0.0f;
    for (long r = r0; r < r1; ++r) {
        float v = h[r * D_OUT + c];            // coalesced across the block
        s += v;
        s2 += v * v;
    }
    unsafeAtomicAdd(&sum[c], s);
    unsafeAtomicAdd(&sumsq[c], s2);
}

// ---------------- fold stats into scale/shift ----------------
__global__ void nm_bn_finalize(const float* __restrict__ sum,
                               const float* __restrict__ sumsq,
                               const float* __restrict__ gamma,
                               const float* __restrict__ beta,
                               float* __restrict__ scale,
                               float* __restrict__ shift) {
    const int c = threadIdx.x;                 // 0..127
    const float invN = 1.0f / (float)N_NODES;
    float mu  = sum[c] * invN;
    float var = sumsq[c] * invN - mu * mu;
    float sc  = gamma[c] * rsqrtf(var + BN_EPS);
    scale[c] = sc;
    shift[c] = beta[c] - sc * mu;
}

// ---------------- apply BN: out = scale[c] * h + shift[c] ----------------
__global__ void nm_bn_apply(const float* __restrict__ h,
                            const float* __restrict__ scale,
                            const float* __restrict__ shift,
                            float* __restrict__ out) {
    const long n4 = (long)N_NODES * D_OUT / 4;
    long i = (long)blockIdx.x * blockDim.x + threadIdx.x;
    long stride = (long)gridDim.x * blockDim.x;
    for (; i < n4; i += stride) {
        int c0 = (int)((i * 4) & (D_OUT - 1));
        float4 v  = ((const float4*)h)[i];
        float4 sc = *(const float4*)(scale + c0);
        float4 sh = *(const float4*)(shift + c0);
        float4 o;
        o.x = v.x * sc.x + sh.x;
        o.y = v.y * sc.y + sh.y;
        o.z = v.z * sc.z + sh.z;
        o.w = v.w * sc.w + sh.w;
        ((float4*)out)[i] = o;
    }
}

extern "C" void kernel_launch(void* const* d_in, const int* in_sizes, int n_in,
                              void* d_out, int out_size, void* d_ws, size_t ws_size,
                              hipStream_t stream) {
    // setup_inputs() order
    const float* edge_attr = (const float*)d_in[0];   // [E, 128]
    const float* node_attr = (const float*)d_in[1];   // [N, 128]
    const float* W         = (const float*)d_in[2];   // [128, 384]
    const float* b         = (const float*)d_in[3];   // [128]
    const float* gamma     = (const float*)d_in[4];   // [128]
    const float* beta      = (const float*)d_in[5];   // [128]
    const int*   row       = (const int*)d_in[6];     // senders
    const int*   col       = (const int*)d_in[7];     // receivers
    float* out = (float*)d_out;

    float* ws     = (float*)d_ws;
    float* recv   = ws + WS_RECV;
    float* send   = ws + WS_SEND;
    float* h      = ws + WS_H;
    float* sum    = ws + WS_SUM;
    float* sumsq  = ws + WS_SUMSQ;
    float* scale  = ws + WS_SCALE;
    float* shift  = ws + WS_SHIFT;

    // 1) zero accumulators (recv+send contiguous) and BN stats
    nm_zero<<<2048, 256, 0, stream>>>(recv, (long)2 * N_NODES * D_IN);
    nm_zero<<<1, 256, 0, stream>>>(sum, 256);   // sum + sumsq

    // 2) scatter-sum edges: E * 32 lanes = 16,000,000 threads
    nm_scatter<<<62500, 256, 0, stream>>>(edge_attr, row, col, recv, send);

    // 3) fp32 WMMA GEMM + bias + ReLU: 625 blocks x 4 waves, 80x32 per wave
    nm_gemm_relu<<<625, 128, 0, stream>>>(recv, send, node_attr, W, b, h);

    // 4) BN batch statistics
    nm_bn_stats<<<(N_NODES + 127) / 128, 128, 0, stream>>>(h, sum, sumsq);
    nm_bn_finalize<<<1, 128, 0, stream>>>(sum, sumsq, gamma, beta, scale, shift);

    // 5) normalize
    nm_bn_apply<<<2048, 256, 0, stream>>>(h, scale, shift, out);
}